// FavorTransAttention_39745627357640
// MI455X (gfx1250) — compile-verified
//
#include <hip/hip_runtime.h>

typedef __attribute__((ext_vector_type(2))) float v2f;
typedef __attribute__((ext_vector_type(8))) float v8f;

#define EPS 1e-3f
#define B_ 8
#define L_ 8192
#define D_ 256

// ---- phase 1: kvs[b][m][d] = sum_l relu(K)[l][m] * V[l][d] ----
#define P1_CHUNK 1024
#define P1_LT    32
#define P1_STR   136              // 128 + 8 pad (TDM pad codes 6/7)
#define P1_BUF   (P1_LT * P1_STR)
#define P1_NC    (P1_CHUNK / P1_LT)

// ---- phase 2: out[l][d] = (q' @ kvs) / (q' @ ksum) ----
#define P2_ROWS 64
#define P2_MC   32
#define P2_STR  268               // 256 + 12 pad (TDM pad codes 7/11)
#define P2_NC   (D_ / P2_MC)

#if __has_builtin(__builtin_amdgcn_tensor_load_to_lds) && __has_builtin(__builtin_amdgcn_s_wait_tensorcnt)
#define HAVE_TDM 1
#else
#define HAVE_TDM 0
#endif

#if HAVE_TDM
typedef unsigned int u32x4 __attribute__((ext_vector_type(4)));
typedef int i32x4 __attribute__((ext_vector_type(4)));
typedef int i32x8 __attribute__((ext_vector_type(8)));

// 2-D tile DMA: global (row-major, stride == tensor_w elems of f32) -> LDS,
// with per-row LDS padding to kill bank conflicts.
__device__ __forceinline__ void tdm_load_2d(const float* gsrc, const float* ldst,
                                            unsigned tensor_w, unsigned tensor_h,
                                            unsigned tile_w, unsigned tile_h,
                                            unsigned pad_interval_code,
                                            unsigned pad_amount_code) {
  unsigned long long ga = (unsigned long long)(const void*)gsrc;
  unsigned lds_off = (unsigned)(unsigned long long)(const void*)ldst; // low 32 = LDS byte offset
  u32x4 g0;
  g0.x = 1u;                                                   // count=1, user mode
  g0.y = lds_off;                                              // lds_addr
  g0.z = (unsigned)ga;                                         // global_addr[31:0]
  g0.w = ((unsigned)(ga >> 32) & 0x01ffffffu) | (2u << 30);    // global_addr[56:32], type=2
  i32x8 g1;
  g1[0] = (int)((2u << 16)                      // data_size = 4B
              | (1u << 20)                      // pad_enable
              | (pad_interval_code << 22)
              | (pad_amount_code << 25));       // wg_mask = 0 (not in cluster)
  g1[1] = (int)((tensor_w & 0xffffu) << 16);                          // tensor_dim0[15:0]
  g1[2] = (int)(((tensor_w >> 16) & 0xffffu) | ((tensor_h & 0xffffu) << 16));
  g1[3] = (int)(((tensor_h >> 16) & 0xffffu) | ((tile_w & 0xffffu) << 16));
  g1[4] = (int)(tile_h & 0xffffu);                                    // tile_dim1 (tile_dim2=0)
  g1[5] = (int)tensor_w;                                              // tensor_dim0_stride
  g1[6] = 0;
  g1[7] = 0;
  i32x4 gz = {0, 0, 0, 0};                                            // 2-D: dims 2+ unused
#if defined(__clang_major__) && (__clang_major__ >= 23)
  i32x8 gz8 = {0, 0, 0, 0, 0, 0, 0, 0};
  __builtin_amdgcn_tensor_load_to_lds(g0, g1, gz, gz, gz8, 0);
#else
  __builtin_amdgcn_tensor_load_to_lds(g0, g1, gz, gz, 0);
#endif
}
#endif

__global__ void performer_zero_ws(float* ws, int n) {
  int i = blockIdx.x * blockDim.x + threadIdx.x;
  for (; i < n; i += gridDim.x * blockDim.x) ws[i] = 0.f;
}

__global__ __launch_bounds__(256) void performer_phase1(
    const float* __restrict__ K, const float* __restrict__ V,
    float* __restrict__ kvs, float* __restrict__ ksum) {
  extern __shared__ float smem[];
  // layout: [buf0 K][buf1 K][buf0 V][buf1 V], each P1_BUF floats

  const int tid  = threadIdx.x;
  const int lane = tid & 31;
  const int w    = tid >> 5;
  const int wm   = w >> 2;              // 0..1 : 64-row m strip
  const int wd   = w & 3;               // 0..3 : 32-col d strip
  const int b    = blockIdx.z;
  const int l_base = blockIdx.y * P1_CHUNK;
  const int mt = blockIdx.x >> 1;
  const int dt = blockIdx.x & 1;
  const int m0 = mt * 128;
  const int d0 = dt * 128;

  const float* Kb = K + (size_t)b * L_ * D_;
  const float* Vb = V + (size_t)b * L_ * D_;

  const v8f zero = {0.f,0.f,0.f,0.f,0.f,0.f,0.f,0.f};
  v8f acc[4][2];
#pragma unroll
  for (int i = 0; i < 4; ++i)
#pragma unroll
    for (int j = 0; j < 2; ++j) acc[i][j] = zero;

  float ksf[4] = {0.f, 0.f, 0.f, 0.f};
  const int kb = (lane >> 4) << 1;
  const int n  = lane & 15;

#if HAVE_TDM
  if (w == 0) {  // prologue: DMA chunk 0 into buffer 0
    tdm_load_2d(&Kb[(size_t)l_base * D_ + m0], smem,              D_, L_, 128, P1_LT, 6, 7);
    tdm_load_2d(&Vb[(size_t)l_base * D_ + d0], smem + 2 * P1_BUF, D_, L_, 128, P1_LT, 6, 7);
  }
#endif

  for (int c = 0; c < P1_NC; ++c) {
    int cur = 0;
#if HAVE_TDM
    cur = c & 1;
    if (w == 0 && (c + 1) < P1_NC) {  // prefetch next chunk into other buffer
      int lrow = l_base + (c + 1) * P1_LT;
      tdm_load_2d(&Kb[(size_t)lrow * D_ + m0], smem + (cur ^ 1) * P1_BUF,
                  D_, L_, 128, P1_LT, 6, 7);
      tdm_load_2d(&Vb[(size_t)lrow * D_ + d0], smem + 2 * P1_BUF + (cur ^ 1) * P1_BUF,
                  D_, L_, 128, P1_LT, 6, 7);
    }
    if (w == 0) {                     // wait current chunk, leave prefetch in flight
      if ((c + 1) < P1_NC) __builtin_amdgcn_s_wait_tensorcnt(2);
      else                 __builtin_amdgcn_s_wait_tensorcnt(0);
    }
    __syncthreads();
#else
    __syncthreads();
#pragma unroll
    for (int i = 0; i < 4; ++i) {
      int idx4 = tid + i * 256;
      int row  = idx4 >> 5;
      int col  = (idx4 & 31) << 2;
      *reinterpret_cast<float4*>(&smem[row * P1_STR + col]) =
          *reinterpret_cast<const float4*>(&Kb[(size_t)(l_base + c * P1_LT + row) * D_ + m0 + col]);
      *reinterpret_cast<float4*>(&smem[2 * P1_BUF + row * P1_STR + col]) =
          *reinterpret_cast<const float4*>(&Vb[(size_t)(l_base + c * P1_LT + row) * D_ + d0 + col]);
    }
    __syncthreads();
#endif

    const float* lK = smem + cur * P1_BUF;
    const float* lV = smem + 2 * P1_BUF + cur * P1_BUF;
    for (int kk = 0; kk < P1_LT; kk += 4) {
      v2f a[4], bb[2];
#pragma unroll
      for (int i = 0; i < 4; ++i) {   // relu feature map applied at frag read
        int base = (kk + kb) * P1_STR + wm * 64 + i * 16 + n;
        a[i].x = fmaxf(lK[base], 0.f) + EPS;
        a[i].y = fmaxf(lK[base + P1_STR], 0.f) + EPS;
        ksf[i] += a[i].x + a[i].y;    // k_sum falls out of the relu'd frags
      }
#pragma unroll
      for (int j = 0; j < 2; ++j) {
        int base = (kk + kb) * P1_STR + wd * 32 + j * 16 + n;
        bb[j].x = lV[base];
        bb[j].y = lV[base + P1_STR];
      }
#pragma unroll
      for (int i = 0; i < 4; ++i)
#pragma unroll
        for (int j = 0; j < 2; ++j)
          acc[i][j] = __builtin_amdgcn_wmma_f32_16x16x4_f32(
              false, a[i], false, bb[j], (short)0, acc[i][j], false, false);
    }
#if HAVE_TDM
    __syncthreads();   // readers done before this buffer is DMA-overwritten
#endif
  }

  const int hi = lane >> 4;
  float* kvsb = kvs + (size_t)b * D_ * D_;
#pragma unroll
  for (int i = 0; i < 4; ++i)
#pragma unroll
    for (int j = 0; j < 2; ++j)
#pragma unroll
      for (int v = 0; v < 8; ++v) {
        int m = m0 + wm * 64 + i * 16 + v + 8 * hi;
        int d = d0 + wd * 32 + j * 16 + n;
        atomicAdd(&kvsb[m * D_ + d], acc[i][j][v]);
      }
  if (dt == 0 && wd == 0) {  // lanes n and n+16 both hold partials of column n
#pragma unroll
    for (int i = 0; i < 4; ++i)
      atomicAdd(&ksum[b * D_ + m0 + wm * 64 + i * 16 + n], ksf[i]);
  }
}

__global__ __launch_bounds__(256) void performer_phase2(
    const float* __restrict__ Q, const float* __restrict__ kvs,
    const float* __restrict__ ksum, float* __restrict__ out) {
  extern __shared__ float smem[];
  // layout: [ldsQ 64*268][ldsB buf0 32*268][ldsB buf1 32*268][ldsS 256][ldsN 64]
  float* ldsQ = smem;
  float* ldsS = smem + P2_ROWS * P2_STR + 2 * P2_MC * P2_STR;
  float* ldsN = ldsS + D_;

  const int tid  = threadIdx.x;
  const int lane = tid & 31;
  const int w    = tid >> 5;
  const int wl   = w >> 1;    // 0..3 : 16-row l strip
  const int wd   = w & 1;     // 0..1 : 128-col d half
  const int b    = blockIdx.y;
  const int l0   = blockIdx.x * P2_ROWS;

  const float* Qb   = Q   + (size_t)b * L_ * D_;
  const float* kvsb = kvs + (size_t)b * D_ * D_;

  if (tid < D_) ldsS[tid] = ksum[b * D_ + tid];

#if HAVE_TDM
  if (w == 0) {  // prologue: Q tile + kvs chunk 0
    tdm_load_2d(&Qb[(size_t)l0 * D_], ldsQ, D_, L_, D_, P2_ROWS, 7, 11);
    tdm_load_2d(kvsb, smem + P2_ROWS * P2_STR, D_, D_, D_, P2_MC, 7, 11);
  }
#else
#pragma unroll
  for (int i = 0; i < 16; ++i) {
    int idx4 = tid + i * 256;
    int row  = idx4 >> 6;
    int col  = (idx4 & 63) << 2;
    *reinterpret_cast<float4*>(&ldsQ[row * P2_STR + col]) =
        *reinterpret_cast<const float4*>(&Qb[(size_t)(l0 + row) * D_ + col]);
  }
#endif

  const v8f zero = {0.f,0.f,0.f,0.f,0.f,0.f,0.f,0.f};
  v8f acc[8];
#pragma unroll
  for (int j = 0; j < 8; ++j) acc[j] = zero;

  const int kb = (lane >> 4) << 1;
  const int n  = lane & 15;

  for (int c = 0; c < P2_NC; ++c) {
    int cur = 0;
#if HAVE_TDM
    cur = c & 1;
    if (w == 0 && (c + 1) < P2_NC)
      tdm_load_2d(&kvsb[(size_t)(c + 1) * P2_MC * D_],
                  smem + P2_ROWS * P2_STR + (cur ^ 1) * P2_MC * P2_STR,
                  D_, D_, D_, P2_MC, 7, 11);
    if (w == 0) {
      if ((c + 1) < P2_NC) __builtin_amdgcn_s_wait_tensorcnt(1);
      else                 __builtin_amdgcn_s_wait_tensorcnt(0);
    }
    __syncthreads();
#else
    __syncthreads();
#pragma unroll
    for (int i = 0; i < 8; ++i) {
      int idx4 = tid + i * 256;
      int row  = idx4 >> 6;
      int col  = (idx4 & 63) << 2;
      *reinterpret_cast<float4*>(&smem[P2_ROWS * P2_STR + row * P2_STR + col]) =
          *reinterpret_cast<const float4*>(&kvsb[(size_t)(c * P2_MC + row) * D_ + col]);
    }
    __syncthreads();
#endif

    if (c == 0 && tid < P2_ROWS) {  // normalizer: sum_m q'[l][m] * ksum[m]
      float s = 0.f;
      for (int m = 0; m < D_; ++m)
        s += (fmaxf(ldsQ[tid * P2_STR + m], 0.f) + EPS) * ldsS[m];
      ldsN[tid] = s;
    }

    const float* lB = smem + P2_ROWS * P2_STR + cur * P2_MC * P2_STR;
    const int mc = c * P2_MC;
    for (int kk = 0; kk < P2_MC; kk += 4) {
      v2f a;
      int abase = (wl * 16 + n) * P2_STR + mc + kk + kb;
      a.x = fmaxf(ldsQ[abase], 0.f) + EPS;      // relu feature map at frag read
      a.y = fmaxf(ldsQ[abase + 1], 0.f) + EPS;
#pragma unroll
      for (int j = 0; j < 8; ++j) {
        int bbase = (kk + kb) * P2_STR + wd * 128 + j * 16 + n;
        v2f bb;
        bb.x = lB[bbase];
        bb.y = lB[bbase + P2_STR];
        acc[j] = __builtin_amdgcn_wmma_f32_16x16x4_f32(
            false, a, false, bb, (short)0, acc[j], false, false);
      }
    }
#if HAVE_TDM
    __syncthreads();
#endif
  }
  __syncthreads();

  const int hi = lane >> 4;
  float rn[8];
#pragma unroll
  for (int v = 0; v < 8; ++v) rn[v] = 1.0f / ldsN[wl * 16 + v + 8 * hi];

  float* outb = out + (size_t)b * L_ * D_;
#pragma unroll
  for (int j = 0; j < 8; ++j)
#pragma unroll
    for (int v = 0; v < 8; ++v) {
      int l = l0 + wl * 16 + v + 8 * hi;
      int d = wd * 128 + j * 16 + n;
      outb[(size_t)l * D_ + d] = acc[j][v] * rn[v];
    }
}

extern "C" void kernel_launch(void* const* d_in, const int* in_sizes, int n_in,
                              void* d_out, int out_size, void* d_ws, size_t ws_size,
                              hipStream_t stream) {
  (void)in_sizes; (void)n_in; (void)out_size; (void)ws_size;
  const float* Q = (const float*)d_in[0];
  const float* K = (const float*)d_in[1];
  const float* V = (const float*)d_in[2];
  float* kvs  = (float*)d_ws;               // B*256*256
  float* ksum = kvs + (size_t)B_ * D_ * D_; // B*256

  int nz = B_ * D_ * D_ + B_ * D_;
  performer_zero_ws<<<256, 256, 0, stream>>>((float*)d_ws, nz);

  size_t lds1 = (size_t)4 * P1_BUF * sizeof(float);  // ~69.6 KB (double-buffered K+V)
  performer_phase1<<<dim3(4, L_ / P1_CHUNK, B_), 256, lds1, stream>>>(K, V, kvs, ksum);

  size_t lds2 = (size_t)(P2_ROWS * P2_STR + 2 * P2_MC * P2_STR + D_ + P2_ROWS) * sizeof(float); // ~138.5 KB
  performer_phase2<<<dim3(L_ / P2_ROWS, B_), 256, lds2, stream>>>(Q, kvs, ksum, (float*)d_out);
}